// AttnGreedySearch_1700807049358
// MI455X (gfx1250) — compile-verified
//
#include <hip/hip_runtime.h>

// CDNA5 / gfx1250: wave32, WMMA f32 16x16x4, Tensor Data Mover staging.
typedef float v2f __attribute__((ext_vector_type(2)));
typedef float v8f __attribute__((ext_vector_type(8)));
typedef unsigned int uint32x4 __attribute__((ext_vector_type(4)));
typedef int int32x4 __attribute__((ext_vector_type(4)));
typedef int int32x8 __attribute__((ext_vector_type(8)));

#define SEARCH 5
#define S_ITEMS 10
#define DIN 100
#define HID 16
#define WPB 8            // waves per block (wave32)
#define TPB (WPB * 32)

#if defined(__HIP_DEVICE_COMPILE__) && __has_builtin(__builtin_amdgcn_tensor_load_to_lds)
#define USE_TDM 1
#else
#define USE_TDM 0
#endif

__launch_bounds__(TPB)
__global__ void attn_greedy_kernel(const float* __restrict__ user_intent,
                                   const float* __restrict__ item_corpus,
                                   const float* __restrict__ W_proj,
                                   const float* __restrict__ b_proj,
                                   const float* __restrict__ W_k,
                                   const float* __restrict__ b_k,
                                   float* __restrict__ out,
                                   int bs)
{
    // Shared weights (block-wide) + per-wave staging buffers.
    __shared__ __align__(16) float sWp[DIN * HID];          // 6400 B
    __shared__ __align__(16) float sWk[HID * HID];          // 1024 B
    __shared__ __align__(16) float sCorp[WPB][16 * DIN];    // 16x100 A-tile per wave
    __shared__ __align__(16) float sTr[WPB][16 * 17];       // padded transpose buffer

    const int tid  = threadIdx.x;
    const int lane = tid & 31;
    const int wave = tid >> 5;
    const int half = lane >> 4;   // lane-half: selects K/M offset in WMMA layouts
    const int lx   = lane & 15;

    for (int i = tid; i < DIN * HID; i += TPB) sWp[i] = W_proj[i];
    for (int i = tid; i < HID * HID; i += TPB) sWk[i] = W_k[i];
    __syncthreads();

    const int b = blockIdx.x * WPB + wave;   // one batch per wave
    if (b < bs) {
        float* corp = &sCorp[wave][0];

        // Zero-pad rows 10..15 only (disjoint from the TDM destination region:
        // tensor ops are unordered vs DS ops, so the overlap must be empty).
        for (int i = S_ITEMS * DIN + lane; i < 16 * DIN; i += 32) corp[i] = 0.0f;

#if USE_TDM
        // ---- Tensor Data Mover: DMA the 10x100 f32 tile global -> LDS.
        // D# group0: count=1 | lds_addr | global_addr[56:0] | type=2.
        // Generic LDS pointer low 32 bits == LDS byte address (flat aperture rule).
        const unsigned long long ga =
            (unsigned long long)(uintptr_t)(item_corpus + (size_t)b * (S_ITEMS * DIN));
        const unsigned lds_addr = (unsigned)(uintptr_t)corp;
        uint32x4 g0 = { 1u,                      // count=1, is_restore=0, gather=0
                        lds_addr,
                        (unsigned)ga,
                        ((unsigned)(ga >> 32) & 0x01FFFFFFu) | (2u << 30) };
        // D# group1: data_size=2 (4B) | tensor_dim0=100 | tensor_dim1=10 |
        //            tile_dim0=100 | tile_dim1=10 | tensor_dim0_stride=100.
        int32x8 g1 = { (int)(2u << 16),          // workgroup_mask=0, data_size=4B
                       (int)(100u << 16),        // tensor_dim0[15:0] @ bits 79:48
                       (int)(10u << 16),         // tensor_dim1[15:0] @ bits 111:80
                       (int)(100u << 16),        // tile_dim0 @ bits 127:112
                       10,                       // tile_dim1 @ bits 143:128
                       100,                      // tensor_dim0_stride lo
                       0, 0 };
        int32x4 gz = { 0, 0, 0, 0 };             // groups 2/3 unused (2D tensor)
#if __clang_major__ >= 23
        int32x8 gz8 = { 0, 0, 0, 0, 0, 0, 0, 0 };
        __builtin_amdgcn_tensor_load_to_lds(g0, g1, gz, gz, gz8, 0);
#else
        __builtin_amdgcn_tensor_load_to_lds(g0, g1, gz, gz, 0);
#endif
        __builtin_amdgcn_s_wait_tensorcnt(0);    // LDS tile visible before DS reads
#else
        // Fallback: coalesced vector-load staging (rows 0..9).
        const float4* g4 = (const float4*)(item_corpus + (size_t)b * (S_ITEMS * DIN));
        float4* l4 = (float4*)corp;
        #pragma unroll
        for (int i = lane; i < (S_ITEMS * DIN) / 4; i += 32) l4[i] = g4[i];
#endif

        // ---- GEMM1: C(16x16) = corpus_tile(16x100) @ W_proj(100x16), 25 k-steps of K=4.
        // A layout (f32 16x4): v0 = {K0 | K2}, v1 = {K1 | K3}, M = lane&15.
        // B layout (f32 4x16): v0 = {K0 | K2}, v1 = {K1 | K3}, N = lane&15.
        v8f c = {};
        #pragma unroll 5
        for (int kk = 0; kk < DIN / 4; ++kk) {
            const int k0 = kk * 4;
            v2f a, bb;
            a.x  = corp[lx * DIN + k0 + 2 * half];
            a.y  = corp[lx * DIN + k0 + 2 * half + 1];
            bb.x = sWp[(k0 + 2 * half) * HID + lx];
            bb.y = sWp[(k0 + 2 * half + 1) * HID + lx];
            c = __builtin_amdgcn_wmma_f32_16x16x4_f32(
                    false, a, false, bb, (short)0, c, false, false);
        }
        const float biasp = b_proj[lx];
        const float biask = b_k[lx];
        #pragma unroll
        for (int v = 0; v < 8; ++v) c[v] += biasp;

        // ---- ui state: running sum + count (src = usum/cnt = mean over appended rows)
        float usum = user_intent[(size_t)b * HID + lx];
        float cntf = 1.0f;
        const size_t obase = (size_t)b * (SEARCH + 1) * HID;
        if (lane < 16) out[obase + lx] = usum;   // ui row 0 = user_intent

        float* tr = &sTr[wave][0];
        for (int t = 1; t <= SEARCH; ++t) {
            // C is in D-layout (M = vgpr slot + 8*half, N = lane&15). To feed it as
            // the A operand of ic@W_k we need the lane-transpose: bounce through LDS
            // (LDS ops are in-order within a wave, no barrier needed).
            #pragma unroll
            for (int v = 0; v < 8; ++v) tr[(v + 8 * half) * 17 + lx] = c[v];

            // newC = C @ W_k + b_k  (K=16 -> 4 WMMA k-steps, accumulate onto bias)
            v8f nc;
            #pragma unroll
            for (int v = 0; v < 8; ++v) nc[v] = biask;
            #pragma unroll
            for (int kc = 0; kc < 4; ++kc) {
                const int k0 = kc * 4;
                v2f a, bb;
                a.x  = tr[lx * 17 + k0 + 2 * half];
                a.y  = tr[lx * 17 + k0 + 2 * half + 1];
                bb.x = sWk[(k0 + 2 * half) * HID + lx];
                bb.y = sWk[(k0 + 2 * half + 1) * HID + lx];
                nc = __builtin_amdgcn_wmma_f32_16x16x4_f32(
                        false, a, false, bb, (short)0, nc, false, false);
            }
            c = nc;

            // ---- score[s] = dot(ic[s,:], usum/cnt); argmax over s<10 (softmax monotone)
            const float srcv = usum / cntf;
            float bestv = -3.4e38f;
            int   besti = 0;
            #pragma unroll
            for (int v = 0; v < 8; ++v) {
                float p = c[v] * srcv;               // row v (half0) / row v+8 (half1)
                p += __shfl_xor(p, 1, 32);           // reduce N within each 16-lane half
                p += __shfl_xor(p, 2, 32);
                p += __shfl_xor(p, 4, 32);
                p += __shfl_xor(p, 8, 32);
                const float q   = __shfl_xor(p, 16, 32);
                const float sv  = (half == 0) ? p : q;   // score[v]
                const float sv8 = (half == 0) ? q : p;   // score[v+8]
                if (sv > bestv)                     { bestv = sv;  besti = v; }
                if (v + 8 < S_ITEMS && sv8 > bestv) { bestv = sv8; besti = v + 8; }
            }

            // ---- gather selected row (wave-uniform besti) to h=lane&15 in all lanes
            float rowreg = c[0];
            #pragma unroll
            for (int v = 1; v < 8; ++v) rowreg = ((besti & 7) == v) ? c[v] : rowreg;
            const float item = __shfl(rowreg, lx | ((besti >> 3) << 4), 32);

            if (lane < 16) out[obase + t * HID + lx] = item;  // append to ui
            usum += item;
            cntf += 1.0f;
        }
    }
}

extern "C" void kernel_launch(void* const* d_in, const int* in_sizes, int n_in,
                              void* d_out, int out_size, void* d_ws, size_t ws_size,
                              hipStream_t stream) {
    const float* user_intent = (const float*)d_in[0];
    const float* item_corpus = (const float*)d_in[1];
    const float* W_proj      = (const float*)d_in[2];
    const float* b_proj      = (const float*)d_in[3];
    const float* W_k         = (const float*)d_in[4];
    const float* b_k         = (const float*)d_in[5];
    float* out = (float*)d_out;

    const int bs = in_sizes[0] / 16;                 // user_intent is (bs, 16)
    const int blocks = (bs + WPB - 1) / WPB;         // one batch per wave32
    attn_greedy_kernel<<<blocks, TPB, 0, stream>>>(
        user_intent, item_corpus, W_proj, b_proj, W_k, b_k, out, bs);
}